// NodeAggregator_3891240370571
// MI455X (gfx1250) — compile-verified
//
#include <hip/hip_runtime.h>
#include <hip/hip_bf16.h>

// Problem sizes (fixed by reference): B=8, N=2048, F=512, K=256
#define BB 8
#define NN 2048
#define FF 512
#define KK 256

typedef float v2f __attribute__((ext_vector_type(2)));
typedef float v8f __attribute__((ext_vector_type(8)));
typedef int   v4i __attribute__((ext_vector_type(4)));

// ---------------------------------------------------------------------------
// CDNA5 async global->LDS copy (ASYNCcnt path), with portable fallback.
// Builtin signature (from hipcc diagnostic): param0 = addrspace(1) v4i*,
// param1 = addrspace(3) v4i*, then imm offset, imm cpol.
// ---------------------------------------------------------------------------
#if __has_builtin(__builtin_amdgcn_global_load_async_to_lds_b128) && \
    __has_builtin(__builtin_amdgcn_s_wait_asynccnt)
#define HAVE_ASYNC 1
#define WAIT_ASYNC(n) __builtin_amdgcn_s_wait_asynccnt(n)
typedef __attribute__((address_space(1))) v4i gv4i;
typedef __attribute__((address_space(3))) v4i lv4i;
#else
#define HAVE_ASYNC 0
#define WAIT_ASYNC(n)
#endif

__device__ __forceinline__ void cp16(const float* g, float* l) {
#if HAVE_ASYNC
  // 16B per lane, global -> LDS, no VGPR round-trip; tracked by ASYNCcnt.
  __builtin_amdgcn_global_load_async_to_lds_b128(
      (gv4i*)g, (lv4i*)l, /*offset=*/0, /*cpol=*/0);
#else
  const float4 v = *reinterpret_cast<const float4*>(g);
  *reinterpret_cast<float4*>(l) = v;
#endif
}

__device__ __forceinline__ v8f wmma4(v2f a, v2f b, v8f c) {
  // V_WMMA_F32_16X16X4_F32: D = A(16x4) * B(4x16) + C(16x16), exact fp32
  return __builtin_amdgcn_wmma_f32_16x16x4_f32(
      /*neg_a=*/false, a, /*neg_b=*/false, b,
      /*c_mod=*/(short)0, c, /*reuse_a=*/false, /*reuse_b=*/false);
}

__device__ __forceinline__ v8f vzero8() {
  v8f z = {0.f, 0.f, 0.f, 0.f, 0.f, 0.f, 0.f, 0.f};
  return z;
}

// ---------------------------------------------------------------------------
// Kernel 1: fused assignment MLP + masked softmax.
// One block = 16 rows of [B*N, .]; 8 waves; wave w owns output columns
// [32w, 32w+32) as two 16-wide WMMA tiles.
// ---------------------------------------------------------------------------
__global__ __launch_bounds__(256)
void assign_softmax_kernel(const float* __restrict__ x, const float* __restrict__ mask,
                           const float* __restrict__ W1, const float* __restrict__ b1,
                           const float* __restrict__ W2, const float* __restrict__ b2,
                           float* __restrict__ S) {
  __shared__ float xs[16][FF + 4];   // x slab; stride 516 floats (2064B, 16B aligned)
  __shared__ float hs[16][KK + 4];   // relu(xW1+b1)
  float (*ls)[FF + 4] = xs;          // logits alias (xs dead after GEMM1)

  const int g0 = blockIdx.x * 16;    // global row base into [B*N]
  const int tid = threadIdx.x;
  const int lane = tid & 31;
  const int w = tid >> 5;

  // Stage 16 x-rows (16 x 512 floats) via async global->LDS B128.
#pragma unroll
  for (int i = 0; i < 8; ++i) {
    int idx = tid + i * 256;         // 0..2047 float4 slots
    int r = idx >> 7;                // 128 float4 per row
    int c4 = (idx & 127) * 4;
    cp16(x + (long long)(g0 + r) * FF + c4, &xs[r][c4]);
  }
  WAIT_ASYNC(0);
  __syncthreads();

  const int m_l = lane & 15;                 // A-frag row (lanes 0-15 == 16-31)
  const int o0  = (lane < 16) ? 0 : 2;       // K sub-offset per lane half
  const int n0  = w * 32 + (lane & 15);      // tile-0 column
  const int n1  = n0 + 16;                   // tile-1 column

  // ---- GEMM1: h = relu(x @ W1 + b1) ----
  v8f acc0 = vzero8(), acc1 = vzero8();
  for (int k = 0; k < FF; k += 4) {
    v2f a   = { xs[m_l][k + o0], xs[m_l][k + o0 + 1] };
    v2f bb0 = { W1[(k + o0) * KK + n0], W1[(k + o0 + 1) * KK + n0] };
    v2f bb1 = { W1[(k + o0) * KK + n1], W1[(k + o0 + 1) * KK + n1] };
    acc0 = wmma4(a, bb0, acc0);
    acc1 = wmma4(a, bb1, acc1);
  }
  float bias0 = b1[n0], bias1 = b1[n1];
#pragma unroll
  for (int r = 0; r < 8; ++r) {
    int row = (lane < 16) ? r : r + 8;       // C/D layout: VGPR r -> M=r / M=8+r
    hs[row][n0] = fmaxf(acc0[r] + bias0, 0.f);
    hs[row][n1] = fmaxf(acc1[r] + bias1, 0.f);
  }
  __syncthreads();

  // ---- GEMM2: logits = h @ W2 + b2 + mask penalty ----
  acc0 = vzero8(); acc1 = vzero8();
  for (int k = 0; k < KK; k += 4) {
    v2f a   = { hs[m_l][k + o0], hs[m_l][k + o0 + 1] };
    v2f bb0 = { W2[(k + o0) * KK + n0], W2[(k + o0 + 1) * KK + n0] };
    v2f bb1 = { W2[(k + o0) * KK + n1], W2[(k + o0 + 1) * KK + n1] };
    acc0 = wmma4(a, bb0, acc0);
    acc1 = wmma4(a, bb1, acc1);
  }
  bias0 = b2[n0]; bias1 = b2[n1];
#pragma unroll
  for (int r = 0; r < 8; ++r) {
    int row = (lane < 16) ? r : r + 8;
    float pen = -1e9f * (1.0f - mask[g0 + row]);
    ls[row][n0] = acc0[r] + bias0 + pen;
    ls[row][n1] = acc1[r] + bias1 + pen;
  }
  __syncthreads();

  // ---- Row softmax over 256 clusters; wave w handles rows 2w, 2w+1 ----
#pragma unroll
  for (int rr = 0; rr < 2; ++rr) {
    int row = w * 2 + rr;
    float vals[8];
    float vmax = -3.4e38f;
#pragma unroll
    for (int i = 0; i < 8; ++i) {
      vals[i] = ls[row][i * 32 + lane];
      vmax = fmaxf(vmax, vals[i]);
    }
#pragma unroll
    for (int off = 16; off >= 1; off >>= 1) vmax = fmaxf(vmax, __shfl_xor(vmax, off, 32));
    float sum = 0.f;
#pragma unroll
    for (int i = 0; i < 8; ++i) { vals[i] = __expf(vals[i] - vmax); sum += vals[i]; }
#pragma unroll
    for (int off = 16; off >= 1; off >>= 1) sum += __shfl_xor(sum, off, 32);
    float inv = 1.0f / sum;
#pragma unroll
    for (int i = 0; i < 8; ++i)
      S[(long long)(g0 + row) * KK + i * 32 + lane] = vals[i] * inv;
  }
}

// ---------------------------------------------------------------------------
// Kernel 2: generic batched fp32 WMMA GEMM.  C[b] = op(A[b]) @ B[b]
// TRANSA=false: A row-major [M,Kc], staged as As[m][k] (direct copy).
// TRANSA=true : A stored [Kc,M], staged as As[k][m]  (direct copy) -> A^T @ B.
// Tiles: BM=BN=64, BK=32; 8 waves, each computing a 16x32 strip (2 wmma tiles).
// Double-buffered async global->LDS staging overlapped with WMMA compute.
// ---------------------------------------------------------------------------
#define BM 64
#define BN 64
#define BKt 32

template <bool TRANSA>
__global__ __launch_bounds__(256)
void gemm_f32_wmma(const float* __restrict__ A, const float* __restrict__ Bm,
                   float* __restrict__ C, int M, int Kc, int N,
                   int lda, int ldb, int ldc,
                   long long sA, long long sB, long long sC) {
  (void)M; (void)N;
  constexpr int AR = TRANSA ? BKt : BM;
  constexpr int AC = (TRANSA ? BM : BKt) + 4;   // +4 floats: bank skew, keeps 16B rows
  __shared__ float As[2][AR][AC];
  __shared__ float Bs[2][BKt][BN + 4];

  const int b = blockIdx.z;
  const float* Ab = A + (long long)b * sA;
  const float* Bb = Bm + (long long)b * sB;
  float* Cb = C + (long long)b * sC;

  const int m0 = blockIdx.x * BM;
  const int n0 = blockIdx.y * BN;
  const int tid = threadIdx.x;
  const int lane = tid & 31;
  const int w = tid >> 5;
  const int wr = w & 3;                       // 4 row-strips of 16
  const int wc = w >> 2;                      // 2 col-strips of 32
  const int m_l = wr * 16 + (lane & 15);
  const int nl0 = wc * 32 + (lane & 15);
  const int nl1 = nl0 + 16;
  const int o0 = (lane < 16) ? 0 : 2;

  // Issue the 4 copy ops (2 A + 2 B float4 slots per thread) for one k-tile.
  auto issue_tile = [&](int buf, int kb) {
#pragma unroll
    for (int i = 0; i < 2; ++i) {
      int idx = tid + i * 256;                // 0..511 float4 slots
      if constexpr (TRANSA) {
        int kk = idx >> 4;                    // 16 float4 per k-row of 64
        int m4 = (idx & 15) * 4;
        cp16(Ab + (long long)(kb + kk) * lda + m0 + m4, &As[buf][kk][m4]);
      } else {
        int m = idx >> 3;                     // 8 float4 per m-row of 32
        int k4 = (idx & 7) * 4;
        cp16(Ab + (long long)(m0 + m) * lda + kb + k4, &As[buf][m][k4]);
      }
    }
#pragma unroll
    for (int i = 0; i < 2; ++i) {
      int idx = tid + i * 256;
      int kk = idx >> 4;
      int n4 = (idx & 15) * 4;
      cp16(Bb + (long long)(kb + kk) * ldb + n0 + n4, &Bs[buf][kk][n4]);
    }
  };

  v8f acc0 = vzero8(), acc1 = vzero8();

  issue_tile(0, 0);                           // prologue
  int cur = 0;
  for (int kb = 0; kb < Kc; kb += BKt) {
    const bool has_next = (kb + BKt) < Kc;
    if (has_next) {
      issue_tile(cur ^ 1, kb + BKt);          // prefetch next tile into other buffer
      WAIT_ASYNC(4);                          // retire current tile (in-order), keep next in flight
    } else {
      WAIT_ASYNC(0);
    }
    __syncthreads();

#pragma unroll
    for (int kk = 0; kk < BKt; kk += 4) {
      float a0, a1;
      if constexpr (TRANSA) {
        a0 = As[cur][kk + o0][m_l];
        a1 = As[cur][kk + o0 + 1][m_l];
      } else {
        a0 = As[cur][m_l][kk + o0];
        a1 = As[cur][m_l][kk + o0 + 1];
      }
      v2f a   = { a0, a1 };
      v2f bb0 = { Bs[cur][kk + o0][nl0], Bs[cur][kk + o0 + 1][nl0] };
      v2f bb1 = { Bs[cur][kk + o0][nl1], Bs[cur][kk + o0 + 1][nl1] };
      acc0 = wmma4(a, bb0, acc0);
      acc1 = wmma4(a, bb1, acc1);
    }
    __syncthreads();                          // everyone done with buf[cur] before overwrite
    cur ^= 1;
  }

#pragma unroll
  for (int r = 0; r < 8; ++r) {
    int rl = (lane < 16) ? r : (r + 8);
    long long row = m0 + wr * 16 + rl;
    Cb[row * ldc + n0 + nl0] = acc0[r];
    Cb[row * ldc + n0 + nl1] = acc1[r];
  }
}

// ---------------------------------------------------------------------------
// Kernel 3: per-row sum -> d_inv_sqrt; also fills pmask with 1.0
// ---------------------------------------------------------------------------
__global__ __launch_bounds__(256)
void rowsum_kernel(const float* __restrict__ pooled, float* __restrict__ dinv,
                   float* __restrict__ pmask) {
  int r = blockIdx.x * blockDim.x + threadIdx.x;  // 0 .. B*K-1
  const float4* row = reinterpret_cast<const float4*>(pooled + (long long)r * KK);
  float s = 0.f;
#pragma unroll 4
  for (int j = 0; j < KK / 4; ++j) {
    float4 v = row[j];
    s += v.x + v.y + v.z + v.w;
  }
  dinv[r] = rsqrtf(s + 1e-9f);
  pmask[r] = 1.0f;
}

// ---------------------------------------------------------------------------
// Kernel 4: pooled_adj[b,i,j] = pooled[b,i,j] * dinv[b,i] * dinv[b,j]
// ---------------------------------------------------------------------------
__global__ __launch_bounds__(256)
void scale_kernel(const float* __restrict__ pooled, const float* __restrict__ dinv,
                  float* __restrict__ out) {
  int idx = blockIdx.x * blockDim.x + threadIdx.x;  // < B*K*K
  int col = idx & (KK - 1);
  int rowg = idx >> 8;                               // b*K + i
  out[idx] = pooled[idx] * dinv[rowg] * dinv[(rowg & ~(KK - 1)) + col];
}

// ---------------------------------------------------------------------------
extern "C" void kernel_launch(void* const* d_in, const int* in_sizes, int n_in,
                              void* d_out, int out_size, void* d_ws, size_t ws_size,
                              hipStream_t stream) {
  (void)in_sizes; (void)n_in; (void)out_size; (void)ws_size;
  const float* x    = (const float*)d_in[0];  // [B,N,F]
  const float* adj  = (const float*)d_in[1];  // [B,N,N]
  const float* mask = (const float*)d_in[2];  // [B,N]
  const float* W1   = (const float*)d_in[3];  // [F,K]
  const float* b1   = (const float*)d_in[4];  // [K]
  const float* W2   = (const float*)d_in[5];  // [K,K]
  const float* b2   = (const float*)d_in[6];  // [K]

  float* out        = (float*)d_out;
  float* pfeat      = out;                                   // [B,K,F]
  float* pooled_adj = out + (long long)BB * KK * FF;         // [B,K,K]
  float* pmask      = pooled_adj + (long long)BB * KK * KK;  // [B,K]

  float* S      = (float*)d_ws;                              // [B*N, K]  16 MB
  float* mid    = S + (long long)BB * NN * KK;               // [B*N, K]  16 MB
  float* pooled = mid + (long long)BB * NN * KK;             // [B,K,K]    2 MB
  float* dinv   = pooled + (long long)BB * KK * KK;          // [B,K]      8 KB

  // 1) S = softmax(relu(x@W1+b1)@W2 + b2 + mask penalty)
  assign_softmax_kernel<<<(BB * NN) / 16, 256, 0, stream>>>(x, mask, W1, b1, W2, b2, S);

  // 2) mid = adj @ S        (M=2048, Kc=2048, N=256 per batch)
  {
    dim3 g(NN / BM, KK / BN, BB);
    gemm_f32_wmma<false><<<g, 256, 0, stream>>>(
        adj, S, mid, NN, NN, KK, NN, KK, KK,
        (long long)NN * NN, (long long)NN * KK, (long long)NN * KK);
  }
  // 3) pooled = S^T @ mid   (M=256, Kc=2048, N=256)
  {
    dim3 g(KK / BM, KK / BN, BB);
    gemm_f32_wmma<true><<<g, 256, 0, stream>>>(
        S, mid, pooled, KK, NN, KK, KK, KK, KK,
        (long long)NN * KK, (long long)NN * KK, (long long)KK * KK);
  }
  // 4) pfeat = S^T @ x      (M=256, Kc=2048, N=512) -> straight into d_out
  {
    dim3 g(KK / BM, FF / BN, BB);
    gemm_f32_wmma<true><<<g, 256, 0, stream>>>(
        S, x, pfeat, KK, NN, FF, KK, FF, FF,
        (long long)NN * KK, (long long)NN * FF, (long long)KK * FF);
  }
  // 5) degree renorm (+ pmask = 1)
  rowsum_kernel<<<(BB * KK) / 256, 256, 0, stream>>>(pooled, dinv, pmask);
  scale_kernel<<<(BB * KK * KK) / 256, 256, 0, stream>>>(pooled, dinv, pooled_adj);
}